// EquivariantEmbedding_72885595013337
// MI455X (gfx1250) — compile-verified
//
#include <hip/hip_runtime.h>
#include <stdint.h>

typedef float v4f __attribute__((ext_vector_type(4)));

#define N_NODES   20000
#define N_EDGES   468000
#define NUM_COEF  16
#define C_CH      128
#define MAX_ELEM  90
#define EMB2_SIZE 32767

__device__ __forceinline__ void nt_store4(float* p, v4f v) {
  __builtin_nontemporal_store(v, (v4f*)p);
}

// ---------------------------------------------------------------------------
// Node embedding: one wave per node. Lane l owns channels [4l, 4l+4).
// Writes 16x128 floats per node (coef 0 = data, coefs 1..15 = zero) with
// nontemporal 128-bit stores (512 B per wave per store instruction).
// ---------------------------------------------------------------------------
__global__ void __launch_bounds__(256)
node_embed_kernel(const float* __restrict__ sphere_w,
                  const float* __restrict__ sphere2_w,
                  const int*   __restrict__ atomic_numbers,
                  const int*   __restrict__ x_bits,
                  float*       __restrict__ out_node) {
  const int lane   = threadIdx.x & 31;
  const int wave   = (blockIdx.x * blockDim.x + threadIdx.x) >> 5;
  const int nwaves = (gridDim.x * blockDim.x) >> 5;

  for (int n = wave; n < N_NODES; n += nwaves) {
    const int an = atomic_numbers[n];
    int idx = 0;
#pragma unroll
    for (int i = 0; i < 15; ++i) idx += x_bits[n * 15 + i] << (14 - i);
    idx = idx < 0 ? 0 : (idx > EMB2_SIZE - 1 ? EMB2_SIZE - 1 : idx);

    const int c = lane * 4;
    v4f a = *(const v4f*)(sphere_w  + (size_t)an  * C_CH + c);
    v4f b = *(const v4f*)(sphere2_w + (size_t)idx * C_CH + c);
    v4f r = a + b;

    float* base = out_node + (size_t)n * (NUM_COEF * C_CH);
    nt_store4(base + c, r);
    v4f z = {0.f, 0.f, 0.f, 0.f};
#pragma unroll
    for (int k = 1; k < NUM_COEF; ++k) nt_store4(base + k * C_CH + c, z);
  }
}

// ---------------------------------------------------------------------------
// Edge kernel: one wave per edge (grid-stride, persistent blocks).
// source_w / target_w (90x128 f32 each, 92 KB total) are staged into LDS once
// per block using gfx1250 async global->LDS copies (ASYNCcnt), then gathered
// per edge with ds_load_b128. Gaussian smearing uses v_exp_f32.
// ---------------------------------------------------------------------------
__global__ void __launch_bounds__(256)
edge_kernel(const float* __restrict__ pos,
            const float* __restrict__ source_w,
            const float* __restrict__ target_w,
            const float* __restrict__ rand_vec,
            const int*   __restrict__ atomic_numbers,
            const int*   __restrict__ edge_index,
            float*       __restrict__ out_edge,
            float*       __restrict__ out_rot) {
  __shared__ __align__(16) float lds_tab[2 * MAX_ELEM * C_CH];  // 92160 bytes

  // ---- stage embedding tables: CDNA5 async global->LDS (ASYNCcnt) ----
  {
    const int tid = threadIdx.x;
    for (int i = tid; i < (MAX_ELEM * C_CH) / 4; i += 256) {
      unsigned gb = (unsigned)(i * 16);
      unsigned l0 = (unsigned)(uintptr_t)(&lds_tab[i * 4]);
      asm volatile("global_load_async_to_lds_b128 %0, %1, %2"
                   :: "v"(l0), "v"(gb), "s"(source_w) : "memory");
      unsigned l1 = (unsigned)(uintptr_t)(&lds_tab[MAX_ELEM * C_CH + i * 4]);
      asm volatile("global_load_async_to_lds_b128 %0, %1, %2"
                   :: "v"(l1), "v"(gb), "s"(target_w) : "memory");
    }
    asm volatile("s_wait_asynccnt 0" ::: "memory");
    __syncthreads();
  }

  const int lane   = threadIdx.x & 31;
  const int wave   = (blockIdx.x * blockDim.x + threadIdx.x) >> 5;
  const int nwaves = (gridDim.x * blockDim.x) >> 5;

  // GaussianSmearing constants: offset_j = j*5/127, coeff = -0.5/(20*delta)^2
  const float DELTA = 5.0f / 127.0f;
  const float COEFF = -0.5f / ((20.0f * DELTA) * (20.0f * DELTA));

  for (int e = wave; e < N_EDGES; e += nwaves) {
    const int src = edge_index[e];
    const int tgt = edge_index[N_EDGES + e];

    // ---- wave-uniform per-edge scalars ----
    const float evx = pos[src * 3 + 0] - pos[tgt * 3 + 0];
    const float evy = pos[src * 3 + 1] - pos[tgt * 3 + 1];
    const float evz = pos[src * 3 + 2] - pos[tgt * 3 + 2];
    const float dist = sqrtf(evx * evx + evy * evy + evz * evz);
    const float di = 1.0f / dist;
    const float nxx = evx * di, nxy = evy * di, nxz = evz * di;

    float e2x = rand_vec[e * 3 + 0] - 0.5f;
    float e2y = rand_vec[e * 3 + 1] - 0.5f;
    float e2z = rand_vec[e * 3 + 2] - 0.5f;
    float rn = 1.0f / sqrtf(e2x * e2x + e2y * e2y + e2z * e2z);
    e2x *= rn; e2y *= rn; e2z *= rn;

    const float bx = -e2y, by = e2x,  bz = e2z;   // rotated fallback 1
    const float cx = e2x,  cy = -e2z, cz = e2y;   // rotated fallback 2
    float vd  = fabsf(e2x * nxx + e2y * nxy + e2z * nxz);
    float vdb = fabsf(bx  * nxx + by  * nxy + bz  * nxz);
    float vdc = fabsf(cx  * nxx + cy  * nxy + cz  * nxz);
    if (vd > vdb) { e2x = bx; e2y = by; e2z = bz; }
    vd = fabsf(e2x * nxx + e2y * nxy + e2z * nxz);
    if (vd > vdc) { e2x = cx; e2y = cy; e2z = cz; }

    // norm_z = normalize(cross(norm_x, e2))
    float nzx = nxy * e2z - nxz * e2y;
    float nzy = nxz * e2x - nxx * e2z;
    float nzz = nxx * e2y - nxy * e2x;
    rn = 1.0f / sqrtf(nzx * nzx + nzy * nzy + nzz * nzz);
    nzx *= rn; nzy *= rn; nzz *= rn;

    // norm_y = normalize(cross(norm_x, norm_z))
    float nyx = nxy * nzz - nxz * nzy;
    float nyy = nxz * nzx - nxx * nzz;
    float nyz = nxx * nzy - nxy * nzx;
    rn = 1.0f / sqrtf(nyx * nyx + nyy * nyy + nyz * nyz);
    nyx *= rn; nyy *= rn; nyz *= rn;

    // ---- edge_feat: [gauss(128) | src_emb(128) | tgt_emb(128)] ----
    float* ef = out_edge + (size_t)e * 384;
    const int ch = lane * 4;

    v4f g;
    {
      float d0 = dist - (float)(ch + 0) * DELTA;
      float d1 = dist - (float)(ch + 1) * DELTA;
      float d2 = dist - (float)(ch + 2) * DELTA;
      float d3 = dist - (float)(ch + 3) * DELTA;
      g.x = __expf(COEFF * d0 * d0);
      g.y = __expf(COEFF * d1 * d1);
      g.z = __expf(COEFF * d2 * d2);
      g.w = __expf(COEFF * d3 * d3);
    }
    nt_store4(ef + ch, g);

    const int an_s = atomic_numbers[src];
    const int an_t = atomic_numbers[tgt];
    v4f se = *(const v4f*)(&lds_tab[an_s * C_CH + ch]);                   // ds_load_b128
    v4f te = *(const v4f*)(&lds_tab[MAX_ELEM * C_CH + an_t * C_CH + ch]); // ds_load_b128
    nt_store4(ef + C_CH + ch, se);
    nt_store4(ef + 2 * C_CH + ch, te);

    // ---- rot matrix: rows [norm_z; norm_x; norm_y], lanes 0..8 each store one ----
    float rv = (lane == 0) ? nzx : (lane == 1) ? nzy : (lane == 2) ? nzz
             : (lane == 3) ? nxx : (lane == 4) ? nxy : (lane == 5) ? nxz
             : (lane == 6) ? nyx : (lane == 7) ? nyy : nyz;
    if (lane < 9) out_rot[(size_t)e * 9 + lane] = rv;
  }
}

extern "C" void kernel_launch(void* const* d_in, const int* in_sizes, int n_in,
                              void* d_out, int out_size, void* d_ws, size_t ws_size,
                              hipStream_t stream) {
  (void)in_sizes; (void)n_in; (void)out_size; (void)d_ws; (void)ws_size;

  const float* pos            = (const float*)d_in[0];
  const float* sphere_w       = (const float*)d_in[1];
  const float* sphere2_w      = (const float*)d_in[2];
  const float* source_w       = (const float*)d_in[3];
  const float* target_w       = (const float*)d_in[4];
  const float* rand_vec       = (const float*)d_in[5];
  const int*   atomic_numbers = (const int*)d_in[6];
  const int*   x_bits         = (const int*)d_in[7];
  const int*   edge_index     = (const int*)d_in[8];

  float* out      = (float*)d_out;
  float* out_node = out;                                                 // [20000,16,128]
  float* out_edge = out + (size_t)N_NODES * NUM_COEF * C_CH;             // [468000,384]
  float* out_rot  = out_edge + (size_t)N_EDGES * 384;                    // [468000,9]

  // One wave per node: 20000 waves = 2500 blocks of 256 threads (8 waves).
  node_embed_kernel<<<2500, 256, 0, stream>>>(sphere_w, sphere2_w,
                                              atomic_numbers, x_bits, out_node);

  // Persistent grid: 1024 blocks x 8 waves; each wave grid-strides ~57 edges,
  // amortizing the 92 KB async LDS table staging per block.
  edge_kernel<<<1024, 256, 0, stream>>>(pos, source_w, target_w, rand_vec,
                                        atomic_numbers, edge_index,
                                        out_edge, out_rot);
}